// BiSGCN_53996328845507
// MI455X (gfx1250) — compile-verified
//
#include <hip/hip_runtime.h>

#define N_NODES 65536
#define N_EDGES 655360
#define IN_C    128
#define OUT_C   64

typedef __attribute__((ext_vector_type(2))) float v2f;
typedef __attribute__((ext_vector_type(8))) float v8f;

// ---- degree / normalization ------------------------------------------------
__global__ void sg_deg_init(float* __restrict__ deg) {
    int i = blockIdx.x * blockDim.x + threadIdx.x;
    deg[i] = 1.0f;                       // self-loop contributes 1
}

__global__ void sg_deg_count(const int* __restrict__ ei, float* __restrict__ deg) {
    int e = blockIdx.x * blockDim.x + threadIdx.x;
    atomicAdd(&deg[ei[N_EDGES + e]], 1.0f);   // dst = edge_index[1][e]
}

__global__ void sg_deg_rsqrt(float* __restrict__ deg) {
    int i = blockIdx.x * blockDim.x + threadIdx.x;
    deg[i] = rsqrtf(deg[i]);             // deg -> dinv in place
}

// ---- hop: initialize output with self-loop term dinv[i]^2 * h_in[i] --------
__global__ void sg_self_init(const float* __restrict__ dinv,
                             const float4* __restrict__ hin,
                             float4* __restrict__ hout) {
    int i = blockIdx.x * blockDim.x + threadIdx.x;   // over N_NODES * 32 float4s
    int node = i >> 5;                                // 32 float4 per node (128 ch)
    float di = dinv[node];
    float w  = di * di;
    float4 v = hin[i];
    float4 r; r.x = w*v.x; r.y = w*v.y; r.z = w*v.z; r.w = w*v.w;
    hout[i] = r;
}

// ---- hop: edge scatter, one wave per edge, lane = 4 channels ---------------
__global__ void sg_scatter(const int* __restrict__ ei,
                           const float* __restrict__ dinv,
                           const float* __restrict__ hin,
                           float* __restrict__ hout) {
    int t    = blockIdx.x * blockDim.x + threadIdx.x;
    int e    = t >> 5;
    int lane = t & 31;
    int s = ei[e];
    int d = ei[N_EDGES + e];
    float w = dinv[s] * dinv[d];
    float4 v = ((const float4*)(hin + (size_t)s * IN_C))[lane];  // coalesced 512B/edge
    float* dst = hout + (size_t)d * IN_C + lane * 4;
    atomicAdd(dst + 0, w * v.x);
    atomicAdd(dst + 1, w * v.y);
    atomicAdd(dst + 2, w * v.z);
    atomicAdd(dst + 3, w * v.w);
}

// ---- terminal GEMM: out[M,64] = h[M,128] @ W[64,128]^T + b, fp32 WMMA ------
// Block: 256 threads = 8 waves; each wave owns one 16x16 output tile.
// Block covers 32 rows x 64 cols. Grid = N_NODES/32.
__global__ void __launch_bounds__(256)
sg_gemm_wmma(const float* __restrict__ h,
             const float* __restrict__ W,
             const float* __restrict__ bias,
             float* __restrict__ out) {
    __shared__ float Wlds[OUT_C * IN_C];             // 32 KB of 320 KB/WGP
    for (int i = threadIdx.x; i < (OUT_C * IN_C) / 4; i += blockDim.x)
        ((float4*)Wlds)[i] = ((const float4*)W)[i];
    __syncthreads();

    const int wave = threadIdx.x >> 5;               // 0..7
    const int lane = threadIdx.x & 31;
    const int m0 = blockIdx.x * 32 + (wave >> 2) * 16;
    const int n0 = (wave & 3) * 16;
    const int lm = lane & 15;                        // row within tile (A) / col (B,D)
    const int kh = (lane >> 4) * 2;                  // K offset: lanes 16-31 hold K+2,K+3

    // A: h rows m0..m0+15, float2 per lane per K-step (16x4 fp32 layout)
    const float* arow = h    + (size_t)(m0 + lm) * IN_C + kh;
    // B: B[k][n] = W[n][k] -> same addressing pattern on W rows (4x16 layout)
    const float* brow = Wlds + (n0 + lm) * IN_C + kh;

    v8f acc = {};
#pragma unroll
    for (int k = 0; k < IN_C; k += 4) {
        v2f a = *(const v2f*)(arow + k);
        v2f b = *(const v2f*)(brow + k);
        // 8 args: (neg_a, A, neg_b, B, c_mod, C, reuse_a, reuse_b)
        acc = __builtin_amdgcn_wmma_f32_16x16x4_f32(
            false, a, false, b, (short)0, acc, false, false);
    }

    float bn = bias[n0 + lm];
    const int mbase = m0 + (lane >> 4) * 8;          // C/D: lanes 16-31 hold M=v+8
#pragma unroll
    for (int v = 0; v < 8; ++v)
        out[(size_t)(mbase + v) * OUT_C + n0 + lm] = acc[v] + bn;
}

// ---- driver ----------------------------------------------------------------
extern "C" void kernel_launch(void* const* d_in, const int* in_sizes, int n_in,
                              void* d_out, int out_size, void* d_ws, size_t ws_size,
                              hipStream_t stream) {
    const float* x    = (const float*)d_in[0];   // [65536,128]
    const int*   ei   = (const int*)  d_in[1];   // [2,655360]
    const float* W    = (const float*)d_in[2];   // [64,128]
    const float* bias = (const float*)d_in[3];   // [64]
    float* out = (float*)d_out;                  // [65536,64]

    float* deg = (float*)d_ws;                   // N floats (becomes dinv)
    float* h1  = deg + N_NODES;                  // N*128 floats
    float* h2  = h1 + (size_t)N_NODES * IN_C;    // N*128 floats

    // normalization
    sg_deg_init <<<N_NODES / 256, 256, 0, stream>>>(deg);
    sg_deg_count<<<N_EDGES / 256, 256, 0, stream>>>(ei, deg);
    sg_deg_rsqrt<<<N_NODES / 256, 256, 0, stream>>>(deg);

    // hop 1: x -> h1
    sg_self_init<<<(N_NODES * 32) / 256, 256, 0, stream>>>(deg, (const float4*)x, (float4*)h1);
    sg_scatter  <<<(N_EDGES * 32) / 256, 256, 0, stream>>>(ei, deg, x, h1);

    // hop 2: h1 -> h2
    sg_self_init<<<(N_NODES * 32) / 256, 256, 0, stream>>>(deg, (const float4*)h1, (float4*)h2);
    sg_scatter  <<<(N_EDGES * 32) / 256, 256, 0, stream>>>(ei, deg, h1, h2);

    // terminal dense linear via fp32 WMMA
    sg_gemm_wmma<<<N_NODES / 32, 256, 0, stream>>>(h2, W, bias, out);
}